// GCN_54958401519774
// MI455X (gfx1250) — compile-verified
//
#include <hip/hip_runtime.h>

#define D 128
#define BN_EPS 1e-5f

typedef __attribute__((ext_vector_type(2))) float v2f;
typedef __attribute__((ext_vector_type(8))) float v8f;

// ---------------- CDNA5 async global->LDS helpers ----------------
// the async copy writes LDS at base + vdst value + offset; generic shared-pointer low 32 bits
// give the workgroup-relative LDS byte address (ISA 10.2 aperture mapping).
__device__ __forceinline__ unsigned lds_byte_addr(const void* p) {
    return (unsigned)(unsigned long long)p;
}

__device__ __forceinline__ void async_g2l_b128(unsigned lds_off, const float4* g) {
    asm volatile("global_load_async_to_lds_b128 %0, %1, off"
                 :: "v"(lds_off), "v"(g)
                 : "memory");
}

__device__ __forceinline__ void wait_async0() {
    asm volatile("s_wait_asynccnt 0x0" ::: "memory");
}

// ---------------- elementwise helpers ----------------

__global__ void k_fill(float* __restrict__ p, float v, int n) {
    int i = blockIdx.x * blockDim.x + threadIdx.x;
    if (i < n) p[i] = v;
}

__global__ void k_deg_count(const int* __restrict__ dstI, float* __restrict__ deg, int E) {
    int e = blockIdx.x * blockDim.x + threadIdx.x;
    if (e < E) atomicAdd(&deg[dstI[e]], 1.0f);
}

__global__ void k_rsqrt_inplace(float* __restrict__ p, int n) {
    int i = blockIdx.x * blockDim.x + threadIdx.x;
    if (i < n) p[i] = rsqrtf(p[i]);   // deg >= 1 always (self-loop)
}

__global__ void k_bias_init(float* __restrict__ out, const float* __restrict__ b, int total) {
    int i = blockIdx.x * blockDim.x + threadIdx.x;
    if (i < total) out[i] = b[i & (D - 1)];
}

// ---------------- WMMA GEMM: C[N,D] = A[N,D] @ W[D,D] ----------------
// Block = 256 threads = 8 waves. Block owns 16 rows; wave w owns cols [16w,16w+16).
// A-tile (16xD, 8KB) staged once; W staged as 32xD K-panels (16KB), both via the
// CDNA5 async global->LDS path (ASYNCcnt) so the copy never touches VGPRs.
// f32 WMMA 16x16x4 lane layout (ISA 7.12.2):
//   A (16x4):  lane L: m=L%16, a[i] = A[m][2*(L/16)+i]
//   B (4x16):  lane L: n=L%16, b[i] = B[2*(L/16)+i][n]
//   C (16x16): lane L: n=L%16, acc[v] = C[v + 8*(L/16)][n]
__global__ __launch_bounds__(256) void k_gemm_wmma(const float* __restrict__ A,
                                                   const float* __restrict__ W,
                                                   float* __restrict__ Cmat, int N) {
    __shared__ float sA[16 * D];   // 8 KB
    __shared__ float sW[32 * D];   // 16 KB

    const int tid  = threadIdx.x;
    const int row0 = blockIdx.x * 16;

    // Async-stage the 16 x D A tile (512 x b128, 2 per thread)
    {
        const float4*  A4  = (const float4*)(A + (size_t)row0 * D);
        const unsigned sAo = lds_byte_addr(sA);
        for (int i = tid; i < 16 * D / 4; i += 256)
            async_g2l_b128(sAo + (unsigned)i * 16u, A4 + i);
    }

    const int wave = tid >> 5;
    const int lane = tid & 31;
    const int half = lane >> 4;    // 0 or 1: selects K pair / row half
    const int mn   = lane & 15;
    const int col0 = wave * 16;
    const unsigned sWo = lds_byte_addr(sW);

    v8f acc = {};
    for (int kp = 0; kp < D; kp += 32) {
        __syncthreads();           // previous panel's LDS reads complete
        // Async-stage W K-panel: rows [kp, kp+32)  (1024 x b128, 4 per thread)
        {
            const float4* W4 = (const float4*)(W + (size_t)kp * D);
            for (int i = tid; i < 32 * D / 4; i += 256)
                async_g2l_b128(sWo + (unsigned)i * 16u, W4 + i);
        }
        wait_async0();             // this wave's async copies landed in LDS
        __syncthreads();           // all waves' copies visible

#pragma unroll
        for (int k2 = 0; k2 < 32; k2 += 4) {
            v2f a, b;
            const int ka = kp + k2 + 2 * half;   // index into full-K A tile
            a.x = sA[mn * D + ka];
            a.y = sA[mn * D + ka + 1];
            const int kb = k2 + 2 * half;        // index into K-panel
            b.x = sW[kb * D + col0 + mn];
            b.y = sW[(kb + 1) * D + col0 + mn];
            acc = __builtin_amdgcn_wmma_f32_16x16x4_f32(
                /*neg_a=*/false, a, /*neg_b=*/false, b,
                /*c_mod=*/(short)0, acc, /*reuse_a=*/false, /*reuse_b=*/false);
        }
    }

#pragma unroll
    for (int v = 0; v < 8; v++) {
        int r = row0 + v + 8 * half;
        if (r < N) Cmat[(size_t)r * D + col0 + mn] = acc[v];
    }
}

// ---------------- edge aggregation: out[dst] += H[src] * dinv[src]*dinv[dst] ----------------
// One wave per (edge or self-loop): 32 lanes x float4 = one 128-float row.
// Feature matrix (25.6 MB) is resident in the 192 MB L2, so the random gather
// and f32 atomic scatter both run at L2 rate; HBM sees each array ~once.
__global__ __launch_bounds__(256) void k_aggregate(const float* __restrict__ H,
                                                   const int* __restrict__ srcI,
                                                   const int* __restrict__ dstI,
                                                   const float* __restrict__ dinv,
                                                   float* __restrict__ out, int E, int N) {
    int wid = blockIdx.x * (blockDim.x >> 5) + (threadIdx.x >> 5);
    if (wid >= E + N) return;
    int lane = threadIdx.x & 31;

    int s, d;
    if (wid < E) { s = srcI[wid]; d = dstI[wid]; }
    else         { s = d = wid - E; }            // self loop

    float nrm = dinv[s] * dinv[d];
    const float4* hv = (const float4*)(H + (size_t)s * D);
    float4 v = hv[lane];
    float* o = out + (size_t)d * D + lane * 4;
    atomicAdd(o + 0, v.x * nrm);
    atomicAdd(o + 1, v.y * nrm);
    atomicAdd(o + 2, v.z * nrm);
    atomicAdd(o + 3, v.w * nrm);
}

// ---------------- BatchNorm stats (per column over N nodes) ----------------
__global__ __launch_bounds__(256) void k_bn_stats(const float* __restrict__ H,
                                                  float* __restrict__ mu,
                                                  float* __restrict__ var, int N) {
    __shared__ float ss[256];
    __shared__ float ss2[256];
    const int d = blockIdx.x;
    const int t = threadIdx.x;
    float s = 0.0f, s2 = 0.0f;
    for (int i = t; i < N; i += 256) {
        float v = H[(size_t)i * D + d];
        s += v; s2 += v * v;
    }
    ss[t] = s; ss2[t] = s2;
    __syncthreads();
    for (int off = 128; off > 0; off >>= 1) {
        if (t < off) { ss[t] += ss[t + off]; ss2[t] += ss2[t + off]; }
        __syncthreads();
    }
    if (t == 0) {
        float m = ss[0] / (float)N;
        mu[d]  = m;
        var[d] = ss2[0] / (float)N - m * m;   // biased variance
    }
}

// ---------------- BatchNorm apply + ReLU (b1 cancels under mean subtraction) ----------------
__global__ void k_bn_relu(float* __restrict__ H, const float* __restrict__ mu,
                          const float* __restrict__ var, const float* __restrict__ gamma,
                          const float* __restrict__ beta, int total) {
    int i = blockIdx.x * blockDim.x + threadIdx.x;
    if (i >= total) return;
    int d = i & (D - 1);
    float v = (H[i] - mu[d]) * rsqrtf(var[d] + BN_EPS) * gamma[d] + beta[d];
    H[i] = v > 0.0f ? v : 0.0f;
}

// ---------------- driver ----------------

extern "C" void kernel_launch(void* const* d_in, const int* in_sizes, int n_in,
                              void* d_out, int out_size, void* d_ws, size_t ws_size,
                              hipStream_t stream) {
    const float* x     = (const float*)d_in[0];
    const int*   ei    = (const int*)d_in[1];   // [2, E] flattened, int32 per harness
    const float* W1    = (const float*)d_in[2];
    // const float* b1 = (const float*)d_in[3];  // cancels under BatchNorm
    const float* gamma = (const float*)d_in[4];
    const float* beta  = (const float*)d_in[5];
    const float* W2    = (const float*)d_in[6];
    const float* b2    = (const float*)d_in[7];

    const int N = in_sizes[0] / D;     // 50000
    const int E = in_sizes[1] / 2;     // 800000
    const int* srcI = ei;
    const int* dstI = ei + E;
    const int ND = N * D;

    // workspace layout (floats)
    float* ws   = (float*)d_ws;
    size_t off  = 0;
    float* dinv = ws + off; off += (size_t)((N + 63) & ~63);
    float* mu   = ws + off; off += D;
    float* var  = ws + off; off += D;
    float* bufA = ws + off; off += (size_t)N * D;
    float* bufB = ws + off;

    float* out = (float*)d_out;

    const int nWave  = E + N;                  // edges + self loops
    const int aggBlk = (nWave + 7) / 8;        // 8 waves per 256-thread block

    // degree -> dinv
    k_fill<<<(N + 255) / 256, 256, 0, stream>>>(dinv, 1.0f, N);          // self loop
    k_deg_count<<<(E + 255) / 256, 256, 0, stream>>>(dstI, dinv, E);
    k_rsqrt_inplace<<<(N + 255) / 256, 256, 0, stream>>>(dinv, N);

    // layer 1: GEMM -> scatter -> BN stats -> BN+ReLU
    k_gemm_wmma<<<(N + 15) / 16, 256, 0, stream>>>(x, W1, bufA, N);
    k_fill<<<(ND + 255) / 256, 256, 0, stream>>>(bufB, 0.0f, ND);
    k_aggregate<<<aggBlk, 256, 0, stream>>>(bufA, srcI, dstI, dinv, bufB, E, N);
    k_bn_stats<<<D, 256, 0, stream>>>(bufB, mu, var, N);
    k_bn_relu<<<(ND + 255) / 256, 256, 0, stream>>>(bufB, mu, var, gamma, beta, ND);

    // layer 2: GEMM -> scatter into (out = b2 broadcast)
    k_gemm_wmma<<<(N + 15) / 16, 256, 0, stream>>>(bufB, W2, bufA, N);
    k_bias_init<<<(ND + 255) / 256, 256, 0, stream>>>(out, b2, ND);
    k_aggregate<<<aggBlk, 256, 0, stream>>>(bufA, srcI, dstI, dinv, out, E, N);
}